// AdaptiveGraphGenerator_82128364634485
// MI455X (gfx1250) — compile-verified
//
#include <hip/hip_runtime.h>
#include <math.h>

#define EPS_F 1e-8f

typedef __attribute__((ext_vector_type(2))) float v2f;
typedef __attribute__((ext_vector_type(8))) float v8f;

// ---------------------------------------------------------------------------
// Kernel 1: per-node weighted norms as a real GEMM via V_WMMA_F32_16X16X4_F32.
//   norm[node][h] = max(sqrt(sum_d X[node][d]^2 * W[h][d]^2), EPS)
// One wave32 computes a 16-node tile: (16x128) x (128x16) with 32 K=4 steps.
// Only columns 0..H-1 of the 16x16 result are meaningful.
//
// A 16x4 f32 layout (ISA 7.12.2): lanes 0-15 hold M=lane, V0=K0,V1=K1;
//                                 lanes 16-31 hold M=lane-16, V0=K2,V1=K3.
// B 4x16 mirrors A's per-VGPR/half K mapping, N = lane&15.
// C 16x16 f32: VGPR j -> M=j (lanes 0-15) / M=j+8 (lanes 16-31), N = lane&15.
// ---------------------------------------------------------------------------
__global__ __launch_bounds__(32) void node_norms_wmma(
    const float* __restrict__ L, const float* __restrict__ R,
    const float* __restrict__ W, float* __restrict__ nl, float* __restrict__ nr,
    int N, int H)
{
    const float* X = (blockIdx.y == 0) ? L : R;
    float* outn    = (blockIdx.y == 0) ? nl : nr;

    const int lane = threadIdx.x;     // 0..31
    const int half = lane >> 4;       // 0 -> K{0,1}, 1 -> K{2,3}
    const int m    = lane & 15;       // A row index / B(N) column index

    const int nodebase = blockIdx.x * 16;
    int node = nodebase + m;
    if (node >= N) node = N - 1;      // keep EXEC all-ones; stores guarded below
    const float* xrow = X + (size_t)node * 128;

    const bool bval = (m < H);
    const float* wrow = W + (size_t)(bval ? m : 0) * 128;

    v8f c = {};
    #pragma unroll 4
    for (int d0 = 0; d0 < 128; d0 += 4) {
        const int d = d0 + 2 * half;
        const float x0 = xrow[d], x1 = xrow[d + 1];
        v2f a; a.x = x0 * x0; a.y = x1 * x1;
        const float w0 = wrow[d], w1 = wrow[d + 1];
        v2f b; b.x = bval ? w0 * w0 : 0.0f;
               b.y = bval ? w1 * w1 : 0.0f;
        // (neg_a, A, neg_b, B, c_mod, C, reuse_a, reuse_b)
        c = __builtin_amdgcn_wmma_f32_16x16x4_f32(false, a, false, b,
                                                  (short)0, c, false, false);
    }

    // Extract columns 0..H-1 (lanes 0..H-1 and 16..16+H-1).
    if (m < H) {
        const int rbase = nodebase + half * 8;
        #pragma unroll
        for (int j = 0; j < 8; ++j) {
            const int rnode = rbase + j;
            if (rnode < N) {
                outn[(size_t)rnode * H + m] = fmaxf(sqrtf(c[j]), EPS_F);
            }
        }
    }
}

// ---------------------------------------------------------------------------
// Kernel 2: edge similarity. One wave per edge iteration; lane covers d-slice
// [4*lane, 4*lane+4) of D=128. L2-gather bound; f32 VALU + wave32 reduction.
// ---------------------------------------------------------------------------
__global__ __launch_bounds__(256) void edge_sim_kernel(
    const float* __restrict__ L, const float* __restrict__ R,
    const int* __restrict__ ei, const float* __restrict__ W,
    const float* __restrict__ thr,
    const float* __restrict__ nl, const float* __restrict__ nr,
    float* __restrict__ out, int E)
{
    const int lane = threadIdx.x & 31;
    const int wid  = blockIdx.x * (blockDim.x >> 5) + (threadIdx.x >> 5);
    const int nw   = gridDim.x * (blockDim.x >> 5);
    const int db   = lane * 4;

    // Per-lane squared metric weights (H=2), loaded once.
    const float4 w0 = *reinterpret_cast<const float4*>(W + db);
    const float4 w1 = *reinterpret_cast<const float4*>(W + 128 + db);
    const float q00 = w0.x*w0.x, q01 = w0.y*w0.y, q02 = w0.z*w0.z, q03 = w0.w*w0.w;
    const float q10 = w1.x*w1.x, q11 = w1.y*w1.y, q12 = w1.z*w1.z, q13 = w1.w*w1.w;
    const float th = 1.0f / (1.0f + expf(-thr[0]));

    for (int e = wid; e < E; e += nw) {
        const int li = ei[e];
        const int ri = ei[E + e];
        const float4 lv = *reinterpret_cast<const float4*>(L + (size_t)li * 128 + db);
        const float4 rv = *reinterpret_cast<const float4*>(R + (size_t)ri * 128 + db);
        const float p0 = lv.x*rv.x, p1 = lv.y*rv.y, p2 = lv.z*rv.z, p3 = lv.w*rv.w;
        float d0 = p0*q00 + p1*q01 + p2*q02 + p3*q03;
        float d1 = p0*q10 + p1*q11 + p2*q12 + p3*q13;
        #pragma unroll
        for (int off = 16; off > 0; off >>= 1) {
            d0 += __shfl_xor(d0, off, 32);
            d1 += __shfl_xor(d1, off, 32);
        }
        if (lane == 0) {
            const float c0 = d0 / (nl[(size_t)li * 2 + 0] * nr[(size_t)ri * 2 + 0]);
            const float c1 = d1 / (nl[(size_t)li * 2 + 1] * nr[(size_t)ri * 2 + 1]);
            const float s = 0.5f * (c0 + c1);
            out[e] = (s < th) ? 0.0f : s;
        }
    }
}

// ---------------------------------------------------------------------------
// Fallback (only if workspace too small): fused kernel, norms computed inline.
// ---------------------------------------------------------------------------
__global__ __launch_bounds__(256) void edge_sim_fused(
    const float* __restrict__ L, const float* __restrict__ R,
    const int* __restrict__ ei, const float* __restrict__ W,
    const float* __restrict__ thr, float* __restrict__ out, int E)
{
    const int lane = threadIdx.x & 31;
    const int wid  = blockIdx.x * (blockDim.x >> 5) + (threadIdx.x >> 5);
    const int nw   = gridDim.x * (blockDim.x >> 5);
    const int db   = lane * 4;

    const float4 w0 = *reinterpret_cast<const float4*>(W + db);
    const float4 w1 = *reinterpret_cast<const float4*>(W + 128 + db);
    const float q00 = w0.x*w0.x, q01 = w0.y*w0.y, q02 = w0.z*w0.z, q03 = w0.w*w0.w;
    const float q10 = w1.x*w1.x, q11 = w1.y*w1.y, q12 = w1.z*w1.z, q13 = w1.w*w1.w;
    const float th = 1.0f / (1.0f + expf(-thr[0]));

    for (int e = wid; e < E; e += nw) {
        const int li = ei[e];
        const int ri = ei[E + e];
        const float4 lv = *reinterpret_cast<const float4*>(L + (size_t)li * 128 + db);
        const float4 rv = *reinterpret_cast<const float4*>(R + (size_t)ri * 128 + db);
        const float p0 = lv.x*rv.x, p1 = lv.y*rv.y, p2 = lv.z*rv.z, p3 = lv.w*rv.w;
        const float l0 = lv.x*lv.x, l1 = lv.y*lv.y, l2 = lv.z*lv.z, l3 = lv.w*lv.w;
        const float r0 = rv.x*rv.x, r1 = rv.y*rv.y, r2 = rv.z*rv.z, r3 = rv.w*rv.w;
        float d0 = p0*q00 + p1*q01 + p2*q02 + p3*q03;
        float d1 = p0*q10 + p1*q11 + p2*q12 + p3*q13;
        float a0 = l0*q00 + l1*q01 + l2*q02 + l3*q03;
        float a1 = l0*q10 + l1*q11 + l2*q12 + l3*q13;
        float b0 = r0*q00 + r1*q01 + r2*q02 + r3*q03;
        float b1 = r0*q10 + r1*q11 + r2*q12 + r3*q13;
        #pragma unroll
        for (int off = 16; off > 0; off >>= 1) {
            d0 += __shfl_xor(d0, off, 32);
            d1 += __shfl_xor(d1, off, 32);
            a0 += __shfl_xor(a0, off, 32);
            a1 += __shfl_xor(a1, off, 32);
            b0 += __shfl_xor(b0, off, 32);
            b1 += __shfl_xor(b1, off, 32);
        }
        if (lane == 0) {
            const float na0 = fmaxf(sqrtf(a0), EPS_F);
            const float na1 = fmaxf(sqrtf(a1), EPS_F);
            const float nb0 = fmaxf(sqrtf(b0), EPS_F);
            const float nb1 = fmaxf(sqrtf(b1), EPS_F);
            const float s = 0.5f * (d0 / (na0 * nb0) + d1 / (na1 * nb1));
            out[e] = (s < th) ? 0.0f : s;
        }
    }
}

extern "C" void kernel_launch(void* const* d_in, const int* in_sizes, int n_in,
                              void* d_out, int out_size, void* d_ws, size_t ws_size,
                              hipStream_t stream) {
    const float* L   = (const float*)d_in[0];
    const float* R   = (const float*)d_in[1];
    const int*   ei  = (const int*)d_in[2];   // (2,E) int32 (JAX x64-off canonicalization)
    const float* W   = (const float*)d_in[3]; // (H,128)
    const float* thr = (const float*)d_in[4]; // (1,)
    float* out = (float*)d_out;

    const int N = in_sizes[0] / 128;
    const int E = in_sizes[2] / 2;
    const int H = in_sizes[3] / 128;          // == 2

    const size_t normBytes = (size_t)N * H * sizeof(float);
    if (ws_size >= 2 * normBytes && H == 2) {
        float* nl = (float*)d_ws;
        float* nr = nl + (size_t)N * H;
        dim3 grid((N + 15) / 16, 2);
        node_norms_wmma<<<grid, 32, 0, stream>>>(L, R, W, nl, nr, N, H);
        edge_sim_kernel<<<4096, 256, 0, stream>>>(L, R, ei, W, thr, nl, nr, out, E);
    } else {
        edge_sim_fused<<<4096, 256, 0, stream>>>(L, R, ei, W, thr, out, E);
    }
}